// CombinedSparsity_7413113552934
// MI455X (gfx1250) — compile-verified
//
#include <hip/hip_runtime.h>
#include <hip/hip_bf16.h>
#include <math.h>

// Problem constants from the reference: activations [128, 128, 64, 64] f32.
#define BATCH 128
#define CHANS 128
#define HW    4096            // 64*64
#define ROWS  (BATCH * CHANS) // 16384 (b,c) planes; row r == b*CHANS + c
#define TOPK  6

#define P1_WAVES 4            // waves (rows) per block in phase 1
#define P1_THREADS (P1_WAVES * 32)

typedef float v4f __attribute__((ext_vector_type(4)));   // native vector: OK for NT builtins

// ---------------------------------------------------------------------------
// Phase 1: per-(b,c) max + first-argmax over 4096 floats.
// One wave per row. The whole 16KB row is DMA'd into LDS with 32 in-flight
// global_load_async_to_lds_b128 ops (CDNA5 async path, ASYNCcnt-tracked),
// then reduced wave-synchronously from LDS.
// ---------------------------------------------------------------------------
__global__ __launch_bounds__(P1_THREADS)
void pool_argmax_kernel(const float* __restrict__ act,
                        float* __restrict__ pooled,
                        int* __restrict__ pidx)
{
    __shared__ __align__(16) float smem[P1_WAVES * HW];   // 64 KB

    const int lane = threadIdx.x & 31;
    const int wave = threadIdx.x >> 5;
    const int r    = blockIdx.x * P1_WAVES + wave;         // plane id

    const unsigned long long rowBase =
        (unsigned long long)(uintptr_t)(act) + (unsigned long long)r * (HW * 4);
    // Low 32 bits of a generic pointer into LDS == byte offset from wave's LDS base.
    const unsigned ldsBase = (unsigned)(uintptr_t)(&smem[wave * HW]);

    // Issue 32 async chunk loads: chunk t moves floats [t*128, t*128+128)
    // (each lane carries 16B; lane's float4 slot = t*32 + lane).
    #pragma unroll
    for (int t = 0; t < 32; ++t) {
        const unsigned byteOff = (unsigned)((t * 32 + lane) * 16);
        const unsigned long long ga = rowBase + byteOff;
        const unsigned           la = ldsBase + byteOff;
        asm volatile("global_load_async_to_lds_b128 %0, %1, off"
                     :: "v"(la), "v"(ga) : "memory");
    }
    // Wait for all async DMA into LDS issued by this wave.
    asm volatile("s_wait_asynccnt 0x0" ::: "memory");

    // Per-lane scan in ascending element order (strict '>' keeps FIRST max,
    // matching jnp.argmax). Lane owns float4 slots t*32+lane -> conflict-free
    // interleaved LDS reads.
    const v4f* tile = reinterpret_cast<const v4f*>(&smem[wave * HW]);
    float best = -INFINITY;
    int   bidx = 0;
    #pragma unroll 4
    for (int t = 0; t < 32; ++t) {
        const int q = t * 32 + lane;
        const v4f v = tile[q];
        const int e = q * 4;
        if (v.x > best) { best = v.x; bidx = e;     }
        if (v.y > best) { best = v.y; bidx = e + 1; }
        if (v.z > best) { best = v.z; bidx = e + 2; }
        if (v.w > best) { best = v.w; bidx = e + 3; }
    }

    // Wave32 cross-lane argmax reduction; tie -> smaller linear index
    // (lanes own disjoint index sets, so this yields the global first-max).
    #pragma unroll
    for (int m = 16; m > 0; m >>= 1) {
        const float ov = __shfl_xor(best, m, 32);
        const int   oi = __shfl_xor(bidx, m, 32);
        if (ov > best || (ov == best && oi < bidx)) { best = ov; bidx = oi; }
    }

    if (lane == 0) {
        pooled[r] = best;
        pidx[r]   = bidx;
    }
}

// ---------------------------------------------------------------------------
// Phase 2: lifetime sparsity. One block per feature c, one thread per batch b.
// Stable rank over the batch column; keep value iff rank < TOPK.
// Exactly reproduces lax.top_k (ties broken toward the lower batch index).
// ---------------------------------------------------------------------------
__global__ __launch_bounds__(BATCH)
void topk_mask_kernel(const float* __restrict__ pooled,
                      float* __restrict__ life)
{
    __shared__ float col[BATCH];
    const int c = blockIdx.x;
    const int b = threadIdx.x;

    const float v = pooled[b * CHANS + c];
    col[b] = v;
    __syncthreads();

    int rank = 0;
    #pragma unroll 8
    for (int j = 0; j < BATCH; ++j) {
        const float u = col[j];
        rank += (u > v) || (u == v && j < b);
    }
    life[b * CHANS + c] = (rank < TOPK) ? v : 0.0f;
}

// ---------------------------------------------------------------------------
// Phase 3: MaxUnpool. One block per (b,c) plane: stream zeros with
// non-temporal b128 stores (256MB write-once -> don't pollute L2), patching
// the argmax element in-register before the store. Single pass.
// ---------------------------------------------------------------------------
__global__ __launch_bounds__(256)
void scatter_fill_kernel(const float* __restrict__ life,
                         const int* __restrict__ pidx,
                         float* __restrict__ out)
{
    const int r = blockIdx.x;                 // plane id == b*CHANS + c
    const float lv = life[r];
    const int   ix = pidx[r];
    float* base = out + (size_t)r * HW;

    #pragma unroll
    for (int it = 0; it < 4; ++it) {          // 256 thr * 4 iters * 4 floats = 4096
        const int e = (it * 256 + (int)threadIdx.x) * 4;
        v4f z = {0.f, 0.f, 0.f, 0.f};
        if (ix >= e && ix < e + 4) {
            z[ix - e] = lv;
        }
        __builtin_nontemporal_store(z, reinterpret_cast<v4f*>(base + e));
    }
}

// ---------------------------------------------------------------------------
extern "C" void kernel_launch(void* const* d_in, const int* in_sizes, int n_in,
                              void* d_out, int out_size, void* d_ws, size_t ws_size,
                              hipStream_t stream)
{
    (void)in_sizes; (void)n_in; (void)out_size; (void)ws_size;

    const float* act = (const float*)d_in[0];
    float* out = (float*)d_out;

    // Workspace layout: pooled[16384] f32 | pidx[16384] i32 | life[16384] f32
    float* pooled = (float*)d_ws;
    int*   pidx   = (int*)(pooled + ROWS);
    float* life   = (float*)(pidx + ROWS);

    pool_argmax_kernel<<<ROWS / P1_WAVES, P1_THREADS, 0, stream>>>(act, pooled, pidx);
    topk_mask_kernel<<<CHANS, BATCH, 0, stream>>>(pooled, life);
    scatter_fill_kernel<<<ROWS, 256, 0, stream>>>(life, pidx, out);
}